// FLAOperator_59631325937748
// MI455X (gfx1250) — compile-verified
//
#include <hip/hip_runtime.h>

#ifndef __has_builtin
#define __has_builtin(x) 0
#endif

typedef __attribute__((ext_vector_type(2))) float v2f;
typedef __attribute__((ext_vector_type(8))) float v8f;
typedef int v4i __attribute__((vector_size(16)));   // matches builtin param type

#define BDIM   2
#define HDIM   16
#define NDIM   4096
#define DDIM   128
#define BH     (BDIM*HDIM)        // 32
#define CHUNKS 32
#define TCHUNK (NDIM/CHUNKS)      // 128 timesteps per block
#define SLAB   16                 // WMMA scan granularity
#define SLABS  (TCHUNK/SLAB)      // 8

// ---- CDNA5 async global->LDS copy (ASYNCcnt path) ------------------------
__device__ __forceinline__ void async_copy16(const void* gsrc, void* ldst) {
#if __has_builtin(__builtin_amdgcn_global_load_async_to_lds_b128)
  __builtin_amdgcn_global_load_async_to_lds_b128(
      (__attribute__((address_space(1))) v4i*)(v4i*)(void*)gsrc,
      (__attribute__((address_space(3))) v4i*)(v4i*)ldst, 0, 0);
#else
  unsigned l = (unsigned)(unsigned long long)
               (__attribute__((address_space(3))) void*)ldst;
  asm volatile("global_load_async_to_lds_b128 %0, %1, off"
               :: "v"(l), "v"(gsrc) : "memory");
#endif
}

__device__ __forceinline__ void wait_async_le8() {
#if __has_builtin(__builtin_amdgcn_s_wait_asynccnt)
  __builtin_amdgcn_s_wait_asynccnt(8);
#else
  asm volatile("s_wait_asynccnt 8" ::: "memory");
#endif
}
__device__ __forceinline__ void wait_async_0() {
#if __has_builtin(__builtin_amdgcn_s_wait_asynccnt)
  __builtin_amdgcn_s_wait_asynccnt(0);
#else
  asm volatile("s_wait_asynccnt 0" ::: "memory");
#endif
}

// fast sigmoid: v_exp_f32 + v_rcp_f32, no IEEE div-fixup chain
__device__ __forceinline__ float fast_sigmoid(float z) {
  return __builtin_amdgcn_rcpf(1.0f + __expf(-z));
}

// ---- Phase 1: per-chunk scan. Intra-slab cumsum via Y = L * X on the ------
// ---- f32 WMMA pipe; inter-slab carry kept per-lane and broadcast. ---------
__global__ __launch_bounds__(256) void gla_phase1(
    const float* __restrict__ q, const float* __restrict__ k,
    const float* __restrict__ v, const float* __restrict__ g,
    float* __restrict__ out, float* __restrict__ totals)
{
  __shared__ float lds[2][4][SLAB*DDIM];   // double-buffered slab, 64 KB

  const int bx    = blockIdx.x;
  const int bh    = bx / CHUNKS;
  const int chunk = bx - bh*CHUNKS;
  const int tid   = threadIdx.x;
  const int lane  = tid & 31;              // wave32
  const int wave  = tid >> 5;              // 8 waves -> 8 d-tiles of 16
  const int hf    = lane >> 4;             // lane half (upper half = M+8 / K+2)
  const int n     = lane & 15;             // column (d) for B/C/D, row (M) for A
  const int d0    = wave * 16;

  const float* srcs[4] = {q, k, v, g};
  const size_t chunk_base = (size_t)bh*NDIM*DDIM + (size_t)chunk*TCHUNK*DDIM;

  // L (16x16 lower-triangular ones, inclusive) as four 16x4 A-layout slices.
  // A layout (32-bit, 16x4): vgpr c, half hf holds K = 4j + 2*hf + c, M = lane&15.
  v2f A[4];
#pragma unroll
  for (int j = 0; j < 4; ++j) {
    A[j].x = ((4*j + 2*hf    ) <= n) ? 1.0f : 0.0f;
    A[j].y = ((4*j + 2*hf + 1) <= n) ? 1.0f : 0.0f;
  }

  // prime slab 0 (each thread: 2 b128 async copies per input array)
#pragma unroll
  for (int a = 0; a < 4; ++a) {
    const float4* gp = (const float4*)(srcs[a] + chunk_base);
    float4* lp = (float4*)(&lds[0][a][0]);
    async_copy16(gp + tid,       lp + tid);
    async_copy16(gp + tid + 256, lp + tid + 256);
  }

  float carry = 0.0f;

  for (int s = 0; s < SLABS; ++s) {
    const int cur = s & 1;
    if (s + 1 < SLABS) {
      const size_t off = chunk_base + (size_t)(s+1)*SLAB*DDIM;
#pragma unroll
      for (int a = 0; a < 4; ++a) {
        const float4* gp = (const float4*)(srcs[a] + off);
        float4* lp = (float4*)(&lds[cur ^ 1][a][0]);
        async_copy16(gp + tid,       lp + tid);
        async_copy16(gp + tid + 256, lp + tid + 256);
      }
      wait_async_le8();                     // slab s done (in-order), s+1 in flight
    } else {
      wait_async_0();
    }
    __syncthreads();

    const float* sq = &lds[cur][0][0];
    const float* sk = &lds[cur][1][0];
    const float* sv = &lds[cur][2][0];
    const float* sg = &lds[cur][3][0];

    v8f acc = {0.f,0.f,0.f,0.f,0.f,0.f,0.f,0.f};
#pragma unroll
    for (int j = 0; j < 4; ++j) {
      // B layout (4x16) mirrors A: vgpr c, half hf holds K = 4j + 2*hf + c, N = lane&15
      const int i0 = (4*j + 2*hf)*DDIM + d0 + n;
      const int i1 = i0 + DDIM;
      const float z0 = sq[i0]*sk[i0] + sg[i0];
      const float z1 = sq[i1]*sk[i1] + sg[i1];
      v2f Bj;
      Bj.x = sv[i0] * fast_sigmoid(z0);
      Bj.y = sv[i1] * fast_sigmoid(z1);
      acc = __builtin_amdgcn_wmma_f32_16x16x4_f32(false, A[j], false, Bj,
                                                  (short)0, acc, false, false);
    }

    // C/D layout: vgpr r holds M = r (lanes 0-15) / M = r+8 (lanes 16-31), N = lane&15
    float y[8];
#pragma unroll
    for (int r = 0; r < 8; ++r) y[r] = acc[r] + carry;

    const size_t rowbase = chunk_base + (size_t)s*SLAB*DDIM
                         + (size_t)(8*hf)*DDIM + (size_t)(d0 + n);
#pragma unroll
    for (int r = 0; r < 8; ++r) out[rowbase + (size_t)r*DDIM] = y[r];

    // new carry = row t=15 value, held in upper-half lanes' y[7]; broadcast it
    carry = __int_as_float(__builtin_amdgcn_ds_bpermute((n + 16) << 2,
                                                        __float_as_int(y[7])));
    __syncthreads();                         // buffer reuse fence
  }

  if (lane < 16)
    totals[(size_t)(bh*CHUNKS + chunk)*DDIM + d0 + n] = carry;
}

// ---- Phase 2: exclusive scan of chunk totals along the chunk axis ---------
__global__ __launch_bounds__(DDIM) void gla_phase2(float* __restrict__ totals)
{
  const int bh = blockIdx.x;
  const int d  = threadIdx.x;
  float run = 0.0f;
  for (int c = 0; c < CHUNKS; ++c) {
    const size_t i = (size_t)(bh*CHUNKS + c)*DDIM + d;
    const float val = totals[i];
    totals[i] = run;
    run += val;
  }
}

// ---- Phase 3: add per-chunk offsets to the locally-scanned output ---------
__global__ __launch_bounds__(256) void gla_phase3(float* __restrict__ out,
                                                  const float* __restrict__ totals)
{
  const long idx4 = (long)blockIdx.x*256 + threadIdx.x;   // float4 index
  const long e  = idx4 * 4;
  const int  d  = (int)(e % DDIM);
  const long r1 = e / DDIM;
  const int  t  = (int)(r1 % NDIM);
  const int  bh = (int)(r1 / NDIM);
  const int  chunk = t / TCHUNK;

  const float4 off = *(const float4*)&totals[(size_t)(bh*CHUNKS + chunk)*DDIM + d];
  float4* ov = (float4*)out;
  float4 o = ov[idx4];
  o.x += off.x; o.y += off.y; o.z += off.z; o.w += off.w;
  ov[idx4] = o;
}

extern "C" void kernel_launch(void* const* d_in, const int* in_sizes, int n_in,
                              void* d_out, int out_size, void* d_ws, size_t ws_size,
                              hipStream_t stream) {
  (void)in_sizes; (void)n_in; (void)out_size; (void)ws_size;
  const float* q = (const float*)d_in[0];
  const float* k = (const float*)d_in[1];
  const float* v = (const float*)d_in[2];
  const float* g = (const float*)d_in[3];
  float* out    = (float*)d_out;
  float* totals = (float*)d_ws;   // BH*CHUNKS*DDIM floats = 512 KB scratch

  gla_phase1<<<BH*CHUNKS, 256, 0, stream>>>(q, k, v, g, out, totals);
  gla_phase2<<<BH, DDIM, 0, stream>>>(totals);
  const int nblk3 = (BDIM*HDIM*NDIM*DDIM) / 4 / 256;      // 16384
  gla_phase3<<<nblk3, 256, 0, stream>>>(out, totals);
}